// Att_H_26474178412691
// MI455X (gfx1250) — compile-verified
//
#include <hip/hip_runtime.h>
#include <hip/hip_bf16.h>
#include <math.h>

// ---------------------------------------------------------------------------
// Types for WMMA f32 16x16x4 (CDNA5 / gfx1250, wave32)
// A: 16x4 f32  -> 2 VGPRs/lane ; B: 4x16 f32 -> 2 VGPRs/lane ; C/D: 8 VGPRs
// ---------------------------------------------------------------------------
typedef __attribute__((ext_vector_type(2))) float v2f;
typedef __attribute__((ext_vector_type(8))) float v8f;

#define B_DIM   512
#define IN_DIM  128
#define H2      512
#define OUT_DIM 25

__device__ __forceinline__ float silu_f(float v) {
    // v * sigmoid(v)
    return v * __builtin_amdgcn_rcpf(1.0f + __expf(-v));
}

// ---------------------------------------------------------------------------
// Register-blocked WMMA GEMM tile body:
//   C[tm:tm+16, tn:tn+32] = silu(X[M,K] * W[N,K]^T + bias[N]) for one wave.
// Two 16x16 output tiles per wave share the A fragment (2 WMMAs per 3 loads).
//
// Per the CDNA5 ISA 32-bit fragment layouts:
//   A (16x4):  lanes 0-15 hold M=lane, K = kk+{0,1}; lanes 16-31 hold K=kk+{2,3}
//   B (4x16):  with B = W^T (W row-major [N,K]), the per-lane load pattern is
//              identical to A's but indexed by the W row (output column).
//   C/D:       element r -> row = r + 8*(lane>=16), col = lane&15
// ---------------------------------------------------------------------------
__device__ __forceinline__ void gemm_tile_16x32(const float* __restrict__ X,
                                                const float* __restrict__ W,
                                                const float* __restrict__ bias,
                                                float* __restrict__ C,
                                                int N, int K,
                                                int tm, int tn, int lane) {
    const int lo = lane & 15;
    const int hi = lane >> 4;              // 0 or 1

    const float* __restrict__ arow  = X + (size_t)(tm + lo) * K + 2 * hi;
    const float* __restrict__ brow0 = W + (size_t)(tn + lo) * K + 2 * hi;
    const float* __restrict__ brow1 = brow0 + (size_t)16 * K;

    v8f acc0 = {};
    v8f acc1 = {};
#pragma unroll 4
    for (int kk = 0; kk < K; kk += 4) {
        v2f a  = *(const v2f*)(arow  + kk);
        v2f b0 = *(const v2f*)(brow0 + kk);
        v2f b1 = *(const v2f*)(brow1 + kk);
        acc0 = __builtin_amdgcn_wmma_f32_16x16x4_f32(
            false, a, false, b0, (short)0, acc0, false, false);
        acc1 = __builtin_amdgcn_wmma_f32_16x16x4_f32(
            false, a, false, b1, (short)0, acc1, false, false);
    }

    const int col0 = tn + lo;
    const int col1 = col0 + 16;
    const float bv0 = bias[col0];
    const float bv1 = bias[col1];
#pragma unroll
    for (int r = 0; r < 8; ++r) {
        const int row = tm + r + 8 * hi;
        C[(size_t)row * N + col0] = silu_f(acc0[r] + bv0);
        C[(size_t)row * N + col1] = silu_f(acc1[r] + bv1);
    }
}

// h = silu(x @ W_in^T + b_in)
__global__ void gemm_silu_wmma(const float* __restrict__ X,
                               const float* __restrict__ W,
                               const float* __restrict__ bias,
                               float* __restrict__ C,
                               int M, int N, int K) {
    const int tiles_n = N >> 5;            // 32-wide tiles
    const int tile    = blockIdx.x * blockDim.y + threadIdx.y;
    const int tm      = (tile / tiles_n) << 4;
    const int tn      = (tile % tiles_n) << 5;
    if (tm >= M) return;
    gemm_tile_16x32(X, W, bias, C, N, K, tm, tn, threadIdx.x);
}

// Q/K/V = silu(h @ {Aq,Ak,Av}^T + {Bq,Bk,Bv}) in a single wide launch
// (blockIdx.y in {0,1,2} selects the head); all three are independent.
__global__ void qkv_gemm_silu_wmma(const float* __restrict__ h,
                                   const float* __restrict__ Aq,
                                   const float* __restrict__ Ak,
                                   const float* __restrict__ Av,
                                   const float* __restrict__ Bq,
                                   const float* __restrict__ Bk,
                                   const float* __restrict__ Bv,
                                   float* __restrict__ Qm,
                                   float* __restrict__ Km,
                                   float* __restrict__ Vm) {
    const float* W    = (blockIdx.y == 0) ? Aq : (blockIdx.y == 1) ? Ak : Av;
    const float* bias = (blockIdx.y == 0) ? Bq : (blockIdx.y == 1) ? Bk : Bv;
    float*       C    = (blockIdx.y == 0) ? Qm : (blockIdx.y == 1) ? Km : Vm;

    const int tiles_n = H2 >> 5;
    const int tile    = blockIdx.x * blockDim.y + threadIdx.y;
    const int tm      = (tile / tiles_n) << 4;
    const int tn      = (tile % tiles_n) << 5;
    gemm_tile_16x32(h, W, bias, C, H2, H2, tm, tn, threadIdx.x);
}

// ---------------------------------------------------------------------------
// Attention: scores[b,i,j] = Q[b,i]*K[b,j]; softmax over j; ctx = silu(attn@V)
// One workgroup per batch row b. K,V staged in LDS as float4 (ds_load_b128).
// Stable softmax: max_j(Q_i*K_j) = Q_i*maxK (Q_i>=0) or Q_i*minK (Q_i<0).
// ---------------------------------------------------------------------------
__global__ void attn_softmax_kernel(const float* __restrict__ Qm,
                                    const float* __restrict__ Km,
                                    const float* __restrict__ Vm,
                                    float* __restrict__ ctx) {
    __shared__ float4 Ks4[H2 / 4];
    __shared__ float4 Vs4[H2 / 4];
    __shared__ float wmax[8], wmin[8];

    const int b    = blockIdx.x;
    const int tid  = threadIdx.x;          // 256 threads = 8 waves
    const int lane = tid & 31;
    const int wave = tid >> 5;

    const float4* __restrict__ Kg = (const float4*)(Km + (size_t)b * H2);
    const float4* __restrict__ Vg = (const float4*)(Vm + (size_t)b * H2);
    if (tid < H2 / 4) {
        Ks4[tid] = Kg[tid];
        Vs4[tid] = Vg[tid];
    }
    __syncthreads();

    // block-wide max/min of K[b,:]
    const float* Ksf = (const float*)Ks4;
    float lmax = -3.4e38f, lmin = 3.4e38f;
    for (int j = tid; j < H2; j += blockDim.x) {
        const float kv = Ksf[j];
        lmax = fmaxf(lmax, kv);
        lmin = fminf(lmin, kv);
    }
#pragma unroll
    for (int off = 16; off > 0; off >>= 1) {
        lmax = fmaxf(lmax, __shfl_xor(lmax, off, 32));
        lmin = fminf(lmin, __shfl_xor(lmin, off, 32));
    }
    if (lane == 0) { wmax[wave] = lmax; wmin[wave] = lmin; }
    __syncthreads();
    float gmax = wmax[0], gmin = wmin[0];
#pragma unroll
    for (int w = 1; w < 8; ++w) {
        gmax = fmaxf(gmax, wmax[w]);
        gmin = fminf(gmin, wmin[w]);
    }

    // each thread covers i = tid and tid+256
    for (int i = tid; i < H2; i += blockDim.x) {
        const float qi = Qm[(size_t)b * H2 + i];
        const float m  = (qi >= 0.0f) ? qi * gmax : qi * gmin;
        float s = 0.0f, sv = 0.0f;
#pragma unroll 4
        for (int j4 = 0; j4 < H2 / 4; ++j4) {
            const float4 kq = Ks4[j4];
            const float4 vq = Vs4[j4];
            float e0 = __expf(fmaf(qi, kq.x, -m));
            float e1 = __expf(fmaf(qi, kq.y, -m));
            float e2 = __expf(fmaf(qi, kq.z, -m));
            float e3 = __expf(fmaf(qi, kq.w, -m));
            s += (e0 + e1) + (e2 + e3);
            sv = fmaf(e0, vq.x, sv);
            sv = fmaf(e1, vq.y, sv);
            sv = fmaf(e2, vq.z, sv);
            sv = fmaf(e3, vq.w, sv);
        }
        ctx[(size_t)b * H2 + i] = silu_f(sv * __builtin_amdgcn_rcpf(s));
    }
}

// ---------------------------------------------------------------------------
// Output head: y = silu(ctx @ W_out^T + b_out) (25 outputs), then the
// per-row quadratic form. One wave per batch row; shfl_xor butterfly
// reductions broadcast each y[o] to all lanes.
//
//   quad = m11*(q0^2+q1^2) + m22*(q2^2+q3^2) + (m12+m21)*(q0*q2+q1*q3)
//   out[b] = quad + mpp
// ---------------------------------------------------------------------------
__global__ void out_head_kernel(const float* __restrict__ ctx,
                                const float* __restrict__ W_out,
                                const float* __restrict__ b_out,
                                float* __restrict__ out) {
    const int b    = blockIdx.x * blockDim.y + threadIdx.y;
    const int lane = threadIdx.x;
    if (b >= B_DIM) return;

    // cache this row of ctx across lanes: lane owns k = lane + 32*t
    float xc[H2 / 32];
#pragma unroll
    for (int t = 0; t < H2 / 32; ++t)
        xc[t] = ctx[(size_t)b * H2 + lane + 32 * t];

    float y[OUT_DIM];
#pragma unroll
    for (int o = 0; o < OUT_DIM; ++o) {
        const float* __restrict__ wrow = W_out + (size_t)o * H2 + lane;
        float acc = 0.0f;
#pragma unroll
        for (int t = 0; t < H2 / 32; ++t)
            acc = fmaf(xc[t], wrow[32 * t], acc);
#pragma unroll
        for (int off = 16; off > 0; off >>= 1)
            acc += __shfl_xor(acc, off, 32);
        y[o] = silu_f(acc + b_out[o]);   // broadcast to all lanes
    }

    if (lane == 0) {
        float m11 = 0.f, m12 = 0.f, m21 = 0.f, m22 = 0.f, mpp = 0.f;
#pragma unroll
        for (int t = 0; t < 5; ++t) {
            m11 += y[t]      * y[t];
            m12 += y[5 + t]  * y[5 + t];
            m21 += y[10 + t] * y[10 + t];
            m22 += y[15 + t] * y[15 + t];
            mpp += y[20 + t] * y[20 + t];
        }
        const float q0 = y[0], q1 = y[1], q2 = y[2], q3 = y[3];
        const float quad = m11 * (q0 * q0 + q1 * q1)
                         + m22 * (q2 * q2 + q3 * q3)
                         + (m12 + m21) * (q0 * q2 + q1 * q3);
        out[b] = quad + mpp;
    }
}

// ---------------------------------------------------------------------------
// Host-side launcher
// ---------------------------------------------------------------------------
extern "C" void kernel_launch(void* const* d_in, const int* in_sizes, int n_in,
                              void* d_out, int out_size, void* d_ws, size_t ws_size,
                              hipStream_t stream) {
    (void)in_sizes; (void)n_in; (void)out_size; (void)ws_size;

    const float* x     = (const float*)d_in[0];
    /* d_in[1] = na (int scalar) — unused by the reference computation */
    const float* W_in  = (const float*)d_in[2];
    const float* b_in  = (const float*)d_in[3];
    const float* Aq    = (const float*)d_in[4];
    const float* Bq    = (const float*)d_in[5];
    const float* Ak    = (const float*)d_in[6];
    const float* Bk    = (const float*)d_in[7];
    const float* Av    = (const float*)d_in[8];
    const float* Bv    = (const float*)d_in[9];
    const float* W_out = (const float*)d_in[10];
    const float* b_out = (const float*)d_in[11];
    float*       out   = (float*)d_out;

    float* ws  = (float*)d_ws;
    float* h   = ws;                         // [512,512]
    float* Qm  = ws + 1 * B_DIM * H2;        // [512,512]
    float* Km  = ws + 2 * B_DIM * H2;        // [512,512]
    float* Vm  = ws + 3 * B_DIM * H2;        // [512,512]
    float* ctx = ws + 4 * B_DIM * H2;        // [512,512]

    const dim3 gblk(32, 4);                  // 4 waves / block, 1 16x32 tile / wave
    const int  tiles   = (B_DIM / 16) * (H2 / 32);   // 512 wave-tiles
    const int  gblocks = tiles / 4;                  // 128 blocks

    // h = silu(x @ W_in^T + b_in)
    gemm_silu_wmma<<<gblocks, gblk, 0, stream>>>(x, W_in, b_in, h,
                                                 B_DIM, H2, IN_DIM);
    // Q/K/V in one wide launch (independent heads on gridDim.y)
    qkv_gemm_silu_wmma<<<dim3(gblocks, 3), gblk, 0, stream>>>(
        h, Aq, Ak, Av, Bq, Bk, Bv, Qm, Km, Vm);

    // scalar-attention softmax + context
    attn_softmax_kernel<<<B_DIM, 256, 0, stream>>>(Qm, Km, Vm, ctx);

    // output head + quadratic reduction
    out_head_kernel<<<B_DIM / 4, dim3(32, 4), 0, stream>>>(ctx, W_out, b_out, out);
}